// Model_42073499632380
// MI455X (gfx1250) — compile-verified
//
#include <hip/hip_runtime.h>
#include <hip/hip_bf16.h>

#define DIM 256
#define T 128
#define RELC_N 26   // REL_COUNT + 1
#define ENT_N 7

typedef float v2f __attribute__((ext_vector_type(2)));
typedef float v8f __attribute__((ext_vector_type(8)));

__device__ __forceinline__ float sigmoidf_(float x) { return 1.0f / (1.0f + __expf(-x)); }

// ---------------------------------------------------------------------------
// Gather word vectors: wvs[t][d] = wordvector[text[t]][d]
// ---------------------------------------------------------------------------
__global__ void gather_kernel(const int* __restrict__ text,
                              const float* __restrict__ wordvector,
                              float* __restrict__ wvs) {
  int t = blockIdx.x;
  int d = threadIdx.x;
  wvs[t * DIM + d] = wordvector[(size_t)text[t] * DIM + d];
}

// ---------------------------------------------------------------------------
// Transpose a submatrix: out[k*R + r] = in[r*ldin + c0 + k],  r<R, k<Kc
// (so serial matvecs read out[] coalesced across threads)
// ---------------------------------------------------------------------------
__global__ void transpose_kernel(const float* __restrict__ in, int ldin, int c0,
                                 float* __restrict__ out, int R, int Kc) {
  int idx = blockIdx.x * blockDim.x + threadIdx.x;
  if (idx >= R * Kc) return;
  int k = idx / R;
  int r = idx - k * R;
  out[idx] = in[(size_t)r * ldin + c0 + k];
}

// ---------------------------------------------------------------------------
// C[M,N] = A[M,K] @ B[:, c0:c0+K].T  (+ bias0 + bias1, per output column)
// A: row-major, lda.  B: weight matrix [N rows, >=c0+K cols], row stride ldb.
// One wave per 16x16 C tile, fp32 WMMA 16x16x4, K stepped 32 (8 WMMA burst).
//
// Padded M rows (arow >= realM): row pointer is CLAMPED in-bounds instead of
// masked — garbage only lands in C rows that the store-guard never writes.
// This keeps EXEC all-1s through the WMMA loop (required) and removes all
// per-iteration exec save/restore.
//
// Fragment layouts (ISA 7.12.2, 32-bit):
//   A 16x4: lanes 0-15 -> M=lane,    K={0,1} in vgpr{0,1}
//           lanes16-31 -> M=lane-16, K={2,3}
//   B 4x16: mirrored (row of B striped across lanes)
//   C:      vgpr r -> M = r + 8*half, N = lane&15
// ---------------------------------------------------------------------------
__global__ __launch_bounds__(128)
void gemm_wmma_f32(const float* __restrict__ A, int lda, int realM,
                   const float* __restrict__ B, int ldb, int c0,
                   float* __restrict__ C, int ldc,
                   const float* __restrict__ bias0,
                   const float* __restrict__ bias1,
                   int Mtiles, int Ntiles, int K) {
  int wave = (blockIdx.x * blockDim.x + threadIdx.x) >> 5;
  int lane = threadIdx.x & 31;
  if (wave >= Mtiles * Ntiles) return;
  int mt = wave / Ntiles;
  int nt = wave - mt * Ntiles;
  int half = lane >> 4;
  int lid  = lane & 15;

  int arow = mt * 16 + lid;
  int srow = (arow < realM) ? arow : (realM - 1);      // clamp, no masking
  const float* Ap = A + (size_t)srow * lda + 2 * half;
  const float* Bp = B + (size_t)(nt * 16 + lid) * ldb + c0 + 2 * half;

  v8f c = {};
  for (int k0 = 0; k0 < K; k0 += 32) {
    v2f a[8], b[8];
#pragma unroll
    for (int u = 0; u < 8; ++u) {                      // 16 b64 loads in flight
      a[u] = *(const v2f*)(Ap + k0 + 4 * u);
      b[u] = *(const v2f*)(Bp + k0 + 4 * u);
    }
#pragma unroll
    for (int u = 0; u < 8; ++u)                        // 8-WMMA burst
      c = __builtin_amdgcn_wmma_f32_16x16x4_f32(false, a[u], false, b[u],
                                                (short)0, c, false, false);
  }

  int col = nt * 16 + lid;
  float badd = 0.f;
  if (bias0) badd += bias0[col];
  if (bias1) badd += bias1[col];
#pragma unroll
  for (int r = 0; r < 8; ++r) {
    int row = mt * 16 + r + 8 * half;
    if (row < realM) C[(size_t)row * ldc + col] = c[r] + badd;
  }
}

// ---------------------------------------------------------------------------
// Bidirectional LSTM recurrence. Xg = x@Wih.T + bih + bhh precomputed ([T,1024]).
// blockIdx.x = 0 -> forward (writes wordin[t*512 + j]),
//            = 1 -> backward (writes wordin[t*512 + 256 + j]).
// 1024 threads: thread r owns gate element r (dot over transposed Whh).
// ---------------------------------------------------------------------------
__global__ __launch_bounds__(1024)
void lstm_kernel(const float* __restrict__ Xg0, const float* __restrict__ Xg1,
                 const float* __restrict__ WT0, const float* __restrict__ WT1,
                 float* __restrict__ wordin) {
  __shared__ float h[DIM], cc[DIM], g[4 * DIM];
  int tid = threadIdx.x;
  bool bwd = (blockIdx.x == 1);
  const float* Xg = bwd ? Xg1 : Xg0;
  const float* WT = bwd ? WT1 : WT0;   // [256 in][1024 out]
  int outOfs = bwd ? DIM : 0;

  if (tid < DIM) { h[tid] = 0.f; cc[tid] = 0.f; }
  __syncthreads();

  for (int s = 0; s < T; ++s) {
    int t = bwd ? (T - 1 - s) : s;
    float acc = Xg[t * 1024 + tid];
#pragma unroll 8
    for (int k = 0; k < DIM; ++k) acc += WT[k * 1024 + tid] * h[k];
    g[tid] = acc;
    __syncthreads();
    if (tid < DIM) {
      float i  = sigmoidf_(g[tid]);
      float f  = sigmoidf_(g[DIM + tid]);
      float gg = tanhf(g[2 * DIM + tid]);
      float o  = sigmoidf_(g[3 * DIM + tid]);
      float cn = f * cc[tid] + i * gg;
      float hn = o * tanhf(cn);
      cc[tid] = cn;
      h[tid]  = hn;
      wordin[t * (2 * DIM) + outOfs + tid] = hn;
    }
    __syncthreads();
  }
}

// ---------------------------------------------------------------------------
// Full top/bot agent scan. One block, 256 threads.
// WTW  [T,256]  = wordin @ Wt[:, :512].T + bt
// RELC [26,256] = relvec @ Wt[:,512:768].T
// WBW  [T,256]  = wordin @ Wb[:, :512].T + bb
// ENTC [7,256]  = entvec @ Wb[:,512:768].T
// *T weights are pre-transposed: W_T[k*256 + j] = W[j][k-slice]
// ---------------------------------------------------------------------------
__global__ __launch_bounds__(256)
void agent_kernel(const float* __restrict__ WTW, const float* __restrict__ WBW,
                  const float* __restrict__ RELCc, const float* __restrict__ ENTCc,
                  const float* __restrict__ WtmT, const float* __restrict__ WttT,
                  const float* __restrict__ WtbT, const float* __restrict__ WbtT,
                  const float* __restrict__ WbmT, const float* __restrict__ WbtgT,
                  const float* __restrict__ Wp,  const float* __restrict__ bp,
                  const float* __restrict__ Wpl, const float* __restrict__ bpl,
                  const float* __restrict__ btt, const float* __restrict__ btb,
                  const float* __restrict__ bbt,
                  float* __restrict__ out) {
  __shared__ float mem[DIM], outp[DIM], target[DIM], memb[DIM], outb[DIM], tgc[DIM];
  __shared__ float logits[32];
  __shared__ int s_i0;
  int j = threadIdx.x;

  float* out_ta = out;                  // top actions   [128]
  float* out_tp = out + T;              // top probs     [128]
  float* out_ba = out + 2 * T;          // bot actions   [128*128]
  float* out_bp = out + 2 * T + T * T;  // bot probs     [128*128]

  mem[j] = 0.f;
  int relAction = 0;  // uniform across threads (updated from shared-broadcast action)
  __syncthreads();

  for (int t = 0; t < T; ++t) {
    // ---- top step: outp = tanh(w-part + rel-part + mem@Wt_mem.T + bt) ----
    float acc = WTW[t * DIM + j] + RELCc[relAction * DIM + j];
#pragma unroll 8
    for (int k = 0; k < DIM; ++k) acc += WtmT[k * DIM + j] * mem[k];
    outp[j] = tanhf(acc);
    __syncthreads();

    if (j < RELC_N) {
      float l = bp[j];
      const float* wr = Wp + j * DIM;
#pragma unroll 8
      for (int k = 0; k < DIM; ++k) l += wr[k] * outp[k];
      logits[j] = l;
    }
    __syncthreads();
    if (j == 0) {
      int am = 0; float mx = logits[0];
      for (int i = 1; i < RELC_N; ++i) if (logits[i] > mx) { mx = logits[i]; am = i; }
      float se = 0.f;
      for (int i = 0; i < RELC_N; ++i) se += __expf(logits[i] - mx);
      s_i0 = am;
      out_ta[t] = (float)am;
      out_tp[t] = 1.f / se;   // softmax value at its own argmax
    }
    __syncthreads();
    int action = s_i0;
    bool do_ner = action > 0;
    int rel_idx = (action > 1 ? action : 1) - 1;
    if (do_ner) relAction = action;

    if (!do_ner) {
      // reference zero-masks the bot rollout and keeps mem = outp: skip it.
      mem[j] = outp[j];
      if (j < T) { out_ba[t * T + j] = 0.f; out_bp[t * T + j] = 0.f; }
      __syncthreads();
      continue;
    }

    // ---- target = outp@Wtt.T + btt ; memb0 = outp@Wtb.T + btb ----
    float ta = btt[j], mb = btb[j];
#pragma unroll 4
    for (int k = 0; k < DIM; ++k) {
      float ov = outp[k];
      ta += WttT[k * DIM + j] * ov;
      mb += WtbT[k * DIM + j] * ov;
    }
    target[j] = ta;
    memb[j] = mb;
    __syncthreads();

    // ---- tgc = target @ Wb[:,1024:1280].T (constant over the bot rollout) ----
    float tg = 0.f;
#pragma unroll 8
    for (int k = 0; k < DIM; ++k) tg += WbtgT[k * DIM + j] * target[k];
    tgc[j] = tg;
    __syncthreads();

    // ---- bot rollout over all words ----
    int ab = 0;
    const float* Wr = Wpl + (size_t)rel_idx * ENT_N * DIM;
    const float* br = bpl + rel_idx * ENT_N;
    for (int w = 0; w < T; ++w) {
      float a2 = WBW[w * DIM + j] + ENTCc[ab * DIM + j] + tgc[j];
#pragma unroll 8
      for (int k = 0; k < DIM; ++k) a2 += WbmT[k * DIM + j] * memb[k];
      outb[j] = tanhf(a2);
      __syncthreads();

      if (j < ENT_N) {
        float l = br[j];
        const float* wrow = Wr + j * DIM;
#pragma unroll 8
        for (int k = 0; k < DIM; ++k) l += wrow[k] * outb[k];
        logits[j] = l;
      }
      __syncthreads();
      if (j == 0) {
        int am = 0; float mx = logits[0];
        for (int i = 1; i < ENT_N; ++i) if (logits[i] > mx) { mx = logits[i]; am = i; }
        float se = 0.f;
        for (int i = 0; i < ENT_N; ++i) se += __expf(logits[i] - mx);
        s_i0 = am;
        out_ba[t * T + w] = (float)am;
        out_bp[t * T + w] = 1.f / se;
      }
      __syncthreads();
      ab = s_i0;
      memb[j] = outb[j];
      __syncthreads();
    }

    // ---- mem_next = membf @ Wbt.T + bbt ----
    float ma = bbt[j];
#pragma unroll 8
    for (int k = 0; k < DIM; ++k) ma += WbtT[k * DIM + j] * memb[k];
    __syncthreads();
    mem[j] = ma;
    __syncthreads();
  }
}

// ---------------------------------------------------------------------------
// Host-side launch
// ---------------------------------------------------------------------------
static inline void launch_transpose(const float* in, int ldin, int c0,
                                    float* out, int R, int Kc, hipStream_t s) {
  int n = R * Kc;
  transpose_kernel<<<(n + 255) / 256, 256, 0, s>>>(in, ldin, c0, out, R, Kc);
}

static inline void launch_gemm(const float* A, int lda, int realM,
                               const float* B, int ldb, int c0,
                               float* C, int ldc,
                               const float* b0, const float* b1,
                               int Mtiles, int Ntiles, int K, hipStream_t s) {
  int tiles = Mtiles * Ntiles;            // one wave per tile, 4 waves/block
  int blocks = (tiles + 3) / 4;
  gemm_wmma_f32<<<blocks, 128, 0, s>>>(A, lda, realM, B, ldb, c0, C, ldc,
                                       b0, b1, Mtiles, Ntiles, K);
}

extern "C" void kernel_launch(void* const* d_in, const int* in_sizes, int n_in,
                              void* d_out, int out_size, void* d_ws, size_t ws_size,
                              hipStream_t stream) {
  const int*   text       = (const int*)  d_in[0];
  const float* wordvector = (const float*)d_in[1];
  const float* relvec     = (const float*)d_in[2];
  const float* entvec     = (const float*)d_in[3];
  const float* WihL = (const float*)d_in[4];
  const float* WhhL = (const float*)d_in[5];
  const float* bihL = (const float*)d_in[6];
  const float* bhhL = (const float*)d_in[7];
  const float* WihR = (const float*)d_in[8];
  const float* WhhR = (const float*)d_in[9];
  const float* bihR = (const float*)d_in[10];
  const float* bhhR = (const float*)d_in[11];
  const float* Wt  = (const float*)d_in[12];  const float* bt  = (const float*)d_in[13];
  const float* Wp  = (const float*)d_in[14];  const float* bp  = (const float*)d_in[15];
  const float* Wb  = (const float*)d_in[16];  const float* bb  = (const float*)d_in[17];
  const float* Wpl = (const float*)d_in[18];  const float* bpl = (const float*)d_in[19];
  const float* Wtt = (const float*)d_in[20];  const float* btt = (const float*)d_in[21];
  const float* Wtb = (const float*)d_in[22];  const float* btb = (const float*)d_in[23];
  const float* Wbt = (const float*)d_in[24];  const float* bbt = (const float*)d_in[25];

  // -------- workspace carve-up (floats) --------
  float* W = (float*)d_ws;
  size_t o = 0;
  float* wvs    = W + o; o += (size_t)T * DIM;          // 128x256
  float* XL     = W + o; o += (size_t)T * 4 * DIM;      // 128x1024
  float* XR     = W + o; o += (size_t)T * 4 * DIM;
  float* wordin = W + o; o += (size_t)T * 2 * DIM;      // 128x512
  float* WTWp   = W + o; o += (size_t)T * DIM;          // 128x256
  float* WBWp   = W + o; o += (size_t)T * DIM;
  float* RELCp  = W + o; o += (size_t)RELC_N * DIM;     // 26x256
  float* ENTCp  = W + o; o += (size_t)ENT_N * DIM;      // 7x256
  float* WhhLT  = W + o; o += (size_t)DIM * 4 * DIM;    // 256x1024
  float* WhhRT  = W + o; o += (size_t)DIM * 4 * DIM;
  float* WtmT   = W + o; o += (size_t)DIM * DIM;        // 256x256 each
  float* WttT   = W + o; o += (size_t)DIM * DIM;
  float* WtbT   = W + o; o += (size_t)DIM * DIM;
  float* WbtT   = W + o; o += (size_t)DIM * DIM;
  float* WbmT   = W + o; o += (size_t)DIM * DIM;
  float* WbtgT  = W + o; o += (size_t)DIM * DIM;
  (void)ws_size; (void)in_sizes; (void)n_in; (void)out_size;

  // -------- stage 1: gather + weight transposes (independent) --------
  gather_kernel<<<T, DIM, 0, stream>>>(text, wordvector, wvs);
  launch_transpose(WhhL, DIM, 0,    WhhLT, 4 * DIM, DIM, stream);
  launch_transpose(WhhR, DIM, 0,    WhhRT, 4 * DIM, DIM, stream);
  launch_transpose(Wt,   4 * DIM, 3 * DIM, WtmT, DIM, DIM, stream); // Wt[:,768:1024]
  launch_transpose(Wtt,  DIM, 0,    WttT,  DIM, DIM, stream);
  launch_transpose(Wtb,  DIM, 0,    WtbT,  DIM, DIM, stream);
  launch_transpose(Wbt,  DIM, 0,    WbtT,  DIM, DIM, stream);
  launch_transpose(Wb,   5 * DIM, 3 * DIM, WbmT,  DIM, DIM, stream); // Wb[:,768:1024]
  launch_transpose(Wb,   5 * DIM, 4 * DIM, WbtgT, DIM, DIM, stream); // Wb[:,1024:1280]

  // -------- stage 2: WMMA GEMM precomputes --------
  // XL/XR = wvs @ Wih.T + (bih + bhh)   [128 x 1024]
  launch_gemm(wvs, DIM, T, WihL, DIM, 0, XL, 4 * DIM, bihL, bhhL, T / 16, (4 * DIM) / 16, DIM, stream);
  launch_gemm(wvs, DIM, T, WihR, DIM, 0, XR, 4 * DIM, bihR, bhhR, T / 16, (4 * DIM) / 16, DIM, stream);
  // RELC = relvec @ Wt[:,512:768].T   [26 x 256]  (M padded to 32, store-guarded)
  launch_gemm(relvec, DIM, RELC_N, Wt, 4 * DIM, 2 * DIM, RELCp, DIM, nullptr, nullptr, 2, DIM / 16, DIM, stream);
  // ENTC = entvec @ Wb[:,512:768].T   [7 x 256]   (M padded to 16, store-guarded)
  launch_gemm(entvec, DIM, ENT_N, Wb, 5 * DIM, 2 * DIM, ENTCp, DIM, nullptr, nullptr, 1, DIM / 16, DIM, stream);

  // -------- stage 3: LSTM recurrences (serial, 2 workgroups) --------
  lstm_kernel<<<2, 1024, 0, stream>>>(XL, XR, WhhLT, WhhRT, wordin);

  // -------- stage 4: wordin-dependent WMMA GEMMs --------
  // WTW = wordin @ Wt[:, :512].T + bt   [128 x 256]
  launch_gemm(wordin, 2 * DIM, T, Wt, 4 * DIM, 0, WTWp, DIM, bt, nullptr, T / 16, DIM / 16, 2 * DIM, stream);
  // WBW = wordin @ Wb[:, :512].T + bb   [128 x 256]
  launch_gemm(wordin, 2 * DIM, T, Wb, 5 * DIM, 0, WBWp, DIM, bb, nullptr, T / 16, DIM / 16, 2 * DIM, stream);

  // -------- stage 5: the serial top/bot agent scan --------
  agent_kernel<<<1, DIM, 0, stream>>>(WTWp, WBWp, RELCp, ENTCp,
                                      WtmT, WttT, WtbT, WbtT, WbmT, WbtgT,
                                      Wp, bp, Wpl, bpl, btt, btb, bbt,
                                      (float*)d_out);
}